// depthwise_patches_41154376631133
// MI455X (gfx1250) — compile-verified
//
#include <hip/hip_runtime.h>
#include <cstdint>

// im2col / extract_patches: x(16,128,128,64) f32, K=3,S=1,D=1
// out(16,126,126,576): out[b,ho,wo,(i*3+j)*64+c] = x[b,ho+i,wo+j,c]
//
// Pure data movement (0 FLOPs). HBM floor ~28us @ 23.3 TB/s (67MB rd + 585MB wr),
// write-bound. Per output row (b,ho): rows ho..ho+2 of the input are contiguous
// (96KB) -> one TDM tensor_load_to_lds issued by wave 0 (TENSORcnt path), then
// all 8 waves stream the 9 shifted patch segments LDS->global with
// global_store_async_from_lds_b128 th:TH_STORE_NT (ASYNCcnt path, non-temporal
// so the 585MB output stream does not evict the L2-resident 67MB input).

#define TPB 256
#define IN_H 128
#define IN_W 128
#define IN_C 64
#define HO 126
#define WO 126
#define ROW_BYTES (IN_W * IN_C * 4)   // 32768 B per input row
#define IN3_BYTES (3 * ROW_BYTES)     // 98304 B staged per workgroup
#define IN3_ELEMS (IN3_BYTES / 4)     // 24576 f32 elements
#define IN3_CHUNKS (IN3_BYTES / 16)   // 6144 x 16B chunks (fallback path)
#define OUT_ROW_CHUNKS (WO * 144)     // 18144 x 16B chunks per output row

typedef __attribute__((ext_vector_type(4))) unsigned int u32x4;
typedef __attribute__((ext_vector_type(8))) int i32x8;
typedef __attribute__((ext_vector_type(4))) int i32x4;

__global__ __launch_bounds__(TPB) void im2col_tdm_kernel(
    const float* __restrict__ x, float* __restrict__ out) {
  extern __shared__ char smem[];  // 96 KB dynamic LDS
  const int tid = threadIdx.x;
  const int wg  = blockIdx.x;     // 0 .. 16*126-1
  const int b   = wg / HO;
  const int ho  = wg - b * HO;

  // Flat LDS pointer: low 32 bits are the workgroup-relative LDS byte offset
  // (ISA 10.2).
  const uint32_t lds_base = (uint32_t)(uintptr_t)smem;

  // ---- Phase 1: global -> LDS, rows ho..ho+2 (one contiguous 96 KB block) ----
  const uint64_t src =
      (uint64_t)(uintptr_t)(x + ((size_t)b * IN_H + ho) * (size_t)(IN_W * IN_C));

#if __has_builtin(__builtin_amdgcn_tensor_load_to_lds)
  if (tid < 32) {  // TDM ops are wave-level; one wave issues the descriptor
    // D# group 0 (ISA 8.3): count=1 | lds_addr | global_addr[56:0] | type=2
    u32x4 g0;
    g0[0] = 1u;                                   // count=1, user descriptor
    g0[1] = lds_base;                             // lds_addr (bytes)
    g0[2] = (uint32_t)src;                        // global_addr[31:0]
    g0[3] = ((uint32_t)(src >> 32) & 0x01FFFFFFu) // global_addr[56:32]
            | 0x80000000u;                        // type=2 ("image")
    // D# group 1 (ISA 8.4): wg_mask=0 (not in cluster), data_size=2 (4B),
    // no barrier/iterate/pad; tensor = 1 x 24576 f32, tile = 1 x 24576,
    // dim0 stride = 24576.
    i32x8 g1;
    g1[0] = 0x00020000;             // [17:16] data_size = 2 (4 bytes)
    g1[1] = (int)((IN3_ELEMS & 0xFFFF) << 16);  // tensor_dim0[15:0] @ bits 63:48
    g1[2] = 0x00010000 |            // tensor_dim1 = 1 @ bits 95:80
            (IN3_ELEMS >> 16);      // tensor_dim0[31:16] @ bits 79:64
    g1[3] = (int)((IN3_ELEMS & 0xFFFF) << 16);  // tile_dim0 @ bits 127:112
    g1[4] = 1;                      // tile_dim1 = 1 @ bits 143:128
    g1[5] = IN3_ELEMS;              // tensor_dim0_stride[31:0] @ bits 191:160
    g1[6] = 0;
    g1[7] = 0;
    i32x4 g2 = {0, 0, 0, 0};        // dims 2/3 unused
    i32x4 g3 = {0, 0, 0, 0};        // dim 4 unused
    i32x8 g4 = {0, 0, 0, 0, 0, 0, 0, 0};  // extra group (6-arg toolchain form)
    __builtin_amdgcn_tensor_load_to_lds(g0, g1, g2, g3, g4, 0);
#if __has_builtin(__builtin_amdgcn_s_wait_tensorcnt)
    __builtin_amdgcn_s_wait_tensorcnt(0);
#else
    asm volatile("s_wait_tensorcnt 0" ::: "memory");
#endif
  }
#else
  // Fallback: per-lane async copies
#pragma unroll
  for (int it = 0; it < IN3_CHUNKS / TPB; ++it) {   // 24 iterations
    const uint32_t chunk = (uint32_t)(it * TPB + tid);
    const uint32_t loff  = lds_base + chunk * 16u;
    const uint64_t gaddr = src + (uint64_t)chunk * 16u;
    asm volatile("global_load_async_to_lds_b128 %0, %1, off"
                 :: "v"(loff), "v"(gaddr)
                 : "memory");
  }
  asm volatile("s_wait_asynccnt 0" ::: "memory");
#endif
  __syncthreads();

  // ---- Phase 2: async LDS -> global, 9 shifted copies, non-temporal ----
  // out float4 chunk f in row: wo = f/144, r = f%144, patch p = r>>4 (i=p/3,
  // j=p%3), cc = r&15. LDS source = i*ROW_BYTES + (wo+j)*256 + cc*16.
  // Output addresses are contiguous across lanes -> 512B per wave store.
  const uint64_t dst = (uint64_t)(uintptr_t)(
      out + ((size_t)b * HO + ho) * (size_t)WO * 576u);
  for (int f = tid; f < OUT_ROW_CHUNKS; f += TPB) {  // 71 iterations
    const int wo = f / 144;
    const int r  = f - wo * 144;
    const int p  = r >> 4;
    const int cc = r & 15;
    const int i  = p / 3;
    const int j  = p - 3 * i;
    const uint32_t loff =
        lds_base + (uint32_t)(i * ROW_BYTES + (wo + j) * (IN_C * 4) + cc * 16);
    const uint64_t gaddr = dst + (uint64_t)f * 16u;
    asm volatile("global_store_async_from_lds_b128 %0, %1, off th:TH_STORE_NT"
                 :: "v"(gaddr), "v"(loff)
                 : "memory");
  }
  // Drain outstanding async stores before LDS is released (kernel end also
  // performs an implicit wait-idle).
  asm volatile("s_wait_asynccnt 0" ::: "memory");
}

extern "C" void kernel_launch(void* const* d_in, const int* in_sizes, int n_in,
                              void* d_out, int out_size, void* d_ws, size_t ws_size,
                              hipStream_t stream) {
  (void)in_sizes; (void)n_in; (void)d_ws; (void)ws_size; (void)out_size;
  const float* x = (const float*)d_in[0];
  float* out = (float*)d_out;
  dim3 grid(16 * HO);   // one workgroup per (b, ho) output row
  dim3 block(TPB);
  im2col_tdm_kernel<<<grid, block, IN3_BYTES, stream>>>(x, out);
}